// NeuralRenderer_10479720202933
// MI455X (gfx1250) — compile-verified
//
#include <hip/hip_runtime.h>

typedef __attribute__((ext_vector_type(2))) float v2f;
typedef __attribute__((ext_vector_type(8))) float v8f;

#define RDIM 256
#define RPIX (RDIM * RDIM)          // 65536 pixels per image
#define RNB 16                      // batch
#define RNC 64                      // circles
#define GROUPS_PER_WAVE 8           // 16-pixel groups per wave
#define WAVES_PER_BLOCK 8
#define PIX_PER_BLOCK (WAVES_PER_BLOCK * GROUPS_PER_WAVE * 16)  // 1024

// D = A(16x4) x B(4x16) + 0 via V_WMMA_F32_16X16X4_F32:
//   A row m  = [px, py, 1, px^2+py^2]
//   B col c  = [-2fu, -2fv, fu^2+fv^2, 1]
//   => D[m][c] = |p - f_c|^2
// Epilogue trick: s = (R^2 - 1e-12) - d; sqrt(s) is qNaN when outside, and
// v_min_num_f32 drops the NaN operand, so no cmp/cndmask is needed at all.
__global__ __launch_bounds__(256) void neural_render_wmma(
    const float* __restrict__ uvd,     // [B][C][3]
    const float* __restrict__ radius,  // [C]
    const int*   __restrict__ dfar_p,  // scalar
    float*       __restrict__ out)     // [B][RPIX]
{
    const int lane = threadIdx.x & 31;
    const int wave = threadIdx.x >> 5;
    const int half = lane >> 4;        // 0: K=0,1 / M=0..7   1: K=2,3 / M=8..15
    const int l15  = lane & 15;

    const int blocksPerImage = RPIX / PIX_PER_BLOCK;   // 64
    const int bb    = blockIdx.x / blocksPerImage;
    const int chunk = blockIdx.x % blocksPerImage;
    const float dfar = (float)(*dfar_p);

    // Per-lane circle parameters for the 4 N-tiles (circle c = 16*t + l15).
    v2f   bmat[4];
    float Dc[4], Rc2e[4];
#pragma unroll
    for (int t = 0; t < 4; ++t) {
        const int c = t * 16 + l15;
        const float* u = uvd + ((size_t)bb * RNC + c) * 3;
        const float fu = u[0], fv = u[1], fd = u[2];
        const float R  = radius[c];
        Dc[t]   = fd;
        Rc2e[t] = R * R - 1e-12f;      // epsilon folded into the threshold
        bmat[t].x = half ? (fu * fu + fv * fv) : (-2.0f * fu);
        bmat[t].y = half ? 1.0f               : (-2.0f * fv);
    }

    const int waveBase = chunk * PIX_PER_BLOCK + wave * (GROUPS_PER_WAVE * 16);
    float* outb = out + (size_t)bb * RPIX;

    // After the folded reduction, this lane holds the finished min for pixel
    // m = r + 8*half with r = {bit3, bit2, bit1} of lane (bit-permuted).
    const int rsel = ((lane >> 3) & 1) | (((lane >> 2) & 1) << 1) | (((lane >> 1) & 1) << 2);
    const int moff = rsel + half * 8;
    const bool writer = (lane & 1) == 0;

#pragma unroll 1
    for (int g = 0; g < GROUPS_PER_WAVE; ++g) {
        const int base = waveBase + g * 16;          // 16-aligned, no x-wrap
        const int x0 = base & (RDIM - 1);
        const int y0 = base >> 8;

        const float px = (float)(x0 + l15);          // M = l15 in both halves
        const float py = (float)y0;
        const float p2 = fmaf(px, px, py * py);
        v2f a;
        a.x = half ? 1.0f : px;
        a.y = half ? p2   : py;

        float minv[8];
#pragma unroll
        for (int r = 0; r < 8; ++r) minv[r] = dfar;

#pragma unroll
        for (int t = 0; t < 4; ++t) {
            v8f cz = {};
            v8f d = __builtin_amdgcn_wmma_f32_16x16x4_f32(
                false, a, false, bmat[t], (short)0, cz, false, false);
            // lane's circle for this tile is c = 16*t + l15 (constant over r)
#pragma unroll
            for (int r = 0; r < 8; ++r) {
                float s     = Rc2e[t] - d[r];                  // >0 <=> inside
                float depth = Dc[t] - __builtin_amdgcn_sqrtf(s); // NaN if outside
                minv[r] = fminf(minv[r], depth);               // min_num drops NaN
            }
        }

        // Folded butterfly min over the 16 circles in each 16-lane half:
        // after each xor step the partials are duplicated, so pack two pixels
        // into complementary lane sub-groups and halve the remaining work.
#pragma unroll
        for (int r = 0; r < 8; ++r)
            minv[r] = fminf(minv[r], __shfl_xor(minv[r], 8, 32));
        float w[4];
#pragma unroll
        for (int j = 0; j < 4; ++j) {
            w[j] = (lane & 8) ? minv[2 * j + 1] : minv[2 * j];
            w[j] = fminf(w[j], __shfl_xor(w[j], 4, 32));
        }
        float u0 = (lane & 4) ? w[1] : w[0];
        float u1 = (lane & 4) ? w[3] : w[2];
        u0 = fminf(u0, __shfl_xor(u0, 2, 32));
        u1 = fminf(u1, __shfl_xor(u1, 2, 32));
        float tv = (lane & 2) ? u1 : u0;
        tv = fminf(tv, __shfl_xor(tv, 1, 32));

        if (writer) outb[base + moff] = tv;
    }
}

extern "C" void kernel_launch(void* const* d_in, const int* in_sizes, int n_in,
                              void* d_out, int out_size, void* d_ws, size_t ws_size,
                              hipStream_t stream) {
    const float* uvd    = (const float*)d_in[0];
    // d_in[1] (UV) is the broadcast pixel grid — regenerated analytically, never read.
    const float* radius = (const float*)d_in[2];
    const int*   dfar   = (const int*)d_in[3];
    float*       out    = (float*)d_out;

    const int grid = RNB * (RPIX / PIX_PER_BLOCK);   // 16 * 64 = 1024 blocks
    neural_render_wmma<<<grid, WAVES_PER_BLOCK * 32, 0, stream>>>(uvd, radius, dfar, out);
}